// FCOS_52390011077066
// MI455X (gfx1250) — compile-verified
//
#include <hip/hip_runtime.h>
#include <cmath>
#include <cstdint>

// FCOS decode, MI455X (gfx1250). Memory-bound (~96 MB traffic, ~65 MFLOP):
// optimize the data path, not math. Async global->LDS staging of the 80-wide
// class block + NT streaming loads/stores.

typedef __attribute__((ext_vector_type(4))) float v4f;
typedef __attribute__((ext_vector_type(4))) int   v4i;

#define AS1 __attribute__((address_space(1)))
#define AS3 __attribute__((address_space(3)))

typedef AS1 v4i* gv4i_p;   // global 16B-vector pointer (builtin arg 0)
typedef AS3 v4i* lv4i_p;   // LDS    16B-vector pointer (builtin arg 1)

#define TILE      128          // pixels per block == threads per block (4 wave32)
#define NCLS      80
#define CPP       20           // 16B chunks per pixel (80 floats / 4)
#define PADF      84           // padded LDS row stride in floats (336 B, kills 8-way conflicts)
#define STRIDE_PX 8.0f

__global__ __launch_bounds__(TILE) void fcos_decode_gfx1250(
    const float* __restrict__ bbox,
    const float* __restrict__ center,
    const float* __restrict__ cls,
    float* __restrict__ out,
    int nPix)
{
    __shared__ float smem[TILE * PADF];          // 43,008 B of the 320 KB WGP LDS

    const int t     = threadIdx.x;
    const int tile0 = blockIdx.x * TILE;         // first pixel of this tile

    // ---- stage the class-logit tile (TILE * 320 B) into LDS via the async
    // ---- copy engine: lane-consecutive 16B chunks -> fully coalesced stream.
    const char* gbase = (const char*)(cls + (size_t)tile0 * NCLS);
#pragma unroll
    for (int i = 0; i < CPP; ++i) {
        int c       = i * TILE + t;              // chunk id, lanes consecutive
        int pix     = c / CPP;
        int k       = c - pix * CPP;
        int ldsByte = pix * (PADF * 4) + k * 16; // padded row layout
        int gByte   = c * 16;
#if __has_builtin(__builtin_amdgcn_global_load_async_to_lds_b128)
        __builtin_amdgcn_global_load_async_to_lds_b128(
            (gv4i_p)(uintptr_t)(gbase + gByte),
            (lv4i_p)((char*)smem + ldsByte),
            /*offset=*/0, /*cpol=*/0);
#else
        unsigned long long ga = (unsigned long long)(uintptr_t)(gbase + gByte);
        unsigned la = (unsigned)(uintptr_t)((char*)smem + ldsByte);
        asm volatile("global_load_async_to_lds_b128 %0, %1, off"
                     :: "v"(la), "v"(ga) : "memory");
#endif
    }

    // Independent global loads overlap the in-flight async copies.
    const int p  = tile0 + t;
    v4f   bb = __builtin_nontemporal_load((const v4f*)bbox + p);
    float ct = __builtin_nontemporal_load(center + p);

    // Own copies complete, then cross-wave visibility via workgroup barrier.
#if __has_builtin(__builtin_amdgcn_s_wait_asynccnt)
    __builtin_amdgcn_s_wait_asynccnt(0);
#else
    asm volatile("s_wait_asynccnt 0x0" ::: "memory");
#endif
    __syncthreads();

    // ---- max/argmax over 80 logits from LDS (first-occurrence semantics) ----
    const v4f* row = (const v4f*)(smem + (size_t)t * PADF);
    float m  = -INFINITY;
    int   mi = 0;
#pragma unroll
    for (int k = 0; k < CPP; ++k) {
        v4f v = row[k];
        if (v.x > m) { m = v.x; mi = 4 * k;     }
        if (v.y > m) { m = v.y; mi = 4 * k + 1; }
        if (v.z > m) { m = v.z; mi = 4 * k + 2; }
        if (v.w > m) { m = v.w; mi = 4 * k + 3; }
    }

    // sigmoid is monotone: max(sigmoid(x)) == sigmoid(max(x)); argmax unchanged.
    float sCls = 1.0f / (1.0f + __expf(-m));
    float sCtr = 1.0f / (1.0f + __expf(-ct));
    float conf = sqrtf(sCls * sCtr);

    float l  = __expf(bb.x) * STRIDE_PX;
    float tp = __expf(bb.y) * STRIDE_PX;
    float r  = __expf(bb.z) * STRIDE_PX;
    float bt = __expf(bb.w) * STRIDE_PX;

    int h = (p >> 7) & 127;                      // nW == nH == 128
    int w = p & 127;
    float cy = (float)h * STRIDE_PX + STRIDE_PX * 0.5f;
    float cx = (float)w * STRIDE_PX + STRIDE_PX * 0.5f;

    v4f o;
    o.x = cx - (l - r) * 0.5f;
    o.y = cy - (tp - bt) * 0.5f;
    o.z = l + r;
    o.w = tp + bt;

    // Outputs: [xywh | cls_idx | conf], all single-touch -> NT stores.
    __builtin_nontemporal_store(o,          (v4f*)out + p);
    __builtin_nontemporal_store((float)mi,  out + (size_t)nPix * 4 + p);
    __builtin_nontemporal_store(conf,       out + (size_t)nPix * 5 + p);
}

extern "C" void kernel_launch(void* const* d_in, const int* in_sizes, int n_in,
                              void* d_out, int out_size, void* d_ws, size_t ws_size,
                              hipStream_t stream)
{
    const float* bbox   = (const float*)d_in[0];   // [16,128,128,4]
    const float* center = (const float*)d_in[1];   // [16,128,128,1]
    const float* cls    = (const float*)d_in[2];   // [16,128,128,80]
    float*       out    = (float*)d_out;           // xywh | idx | conf concatenated

    const int nPix = in_sizes[1];                  // 16*128*128 = 262144, TILE-divisible
    const int grid = nPix / TILE;                  // 2048 blocks of 128 threads

    hipLaunchKernelGGL(fcos_decode_gfx1250, dim3(grid), dim3(TILE), 0, stream,
                       bbox, center, cls, out, nPix);
}